// ConceptEmbedding_5050881540380
// MI455X (gfx1250) — compile-verified
//
#include <hip/hip_runtime.h>

// Fused type-conditioned embedding gather for MI455X (gfx1250, wave32).
//
// out[b,s,:] = (t==1)*proc[c] + (t==2)*med[c] + (t==3)*chart[c], zeros otherwise
//
// Memory-bound: ~16 MB NT write + ~12 MB gather => ~1.2 us at 23.3 TB/s.
//
// Branchless design:
//  * One wave32 per token row: E=128 f32 = 512 B = 32 lanes x float4.
//    One global_load_b128 + one NT global_store_b128 per lane, fully coalesced.
//  * token_type/concept are wave-uniform -> readfirstlane pins them to SGPRs;
//    table base chosen with scalar cselects (no branches).
//  * Data invariant from the reference: row PADDING_IDX==1 of every table is
//    all-zeros, so the t==0 ("write zeros") case is just an unconditional
//    gather of row 1 -> straight-line code, no execz paths.
//  * Tables total 153.6 MB < 192 MB L2: gathers stay regular-temporal so they
//    become L2 hits on graph replay; the write-once output uses a NON-TEMPORAL
//    store so it does not evict the tables.

typedef __attribute__((ext_vector_type(4))) float v4f;

#define EMB_DIM  128
#define N_TOKENS (16 * 2048)   // B * S
#define PAD_ROW  1             // zeroed padding row in every table

__global__ __launch_bounds__(256) void concept_embed_gather(
    const float* __restrict__ proc_emb,
    const float* __restrict__ med_emb,
    const float* __restrict__ chart_emb,
    const int*  __restrict__ concept_ids,
    const int*  __restrict__ token_type,
    float* __restrict__ out)
{
    const unsigned tid  = blockIdx.x * blockDim.x + threadIdx.x;
    const unsigned tok  = tid >> 5;   // one wave32 per token row
    const unsigned lane = tid & 31;   // one float4 (16 B) per lane -> 512 B row

    // Wave-uniform scalars -> SGPRs (saddr-form b128 gather, scalar selects).
    const int t   = __builtin_amdgcn_readfirstlane(token_type[tok]);
    const int idx = __builtin_amdgcn_readfirstlane(concept_ids[tok]);

    // Branchless table select; invalid types read the all-zero padding row.
    const float* base = (t == 2) ? med_emb : ((t == 3) ? chart_emb : proc_emb);
    const int    row  = (t >= 1 && t <= 3) ? idx : PAD_ROW;

    const v4f* src = (const v4f*)(base + (size_t)row * EMB_DIM);
    const v4f  v   = src[lane];

    v4f* dst = (v4f*)(out + (size_t)tok * EMB_DIM) + lane;
    __builtin_nontemporal_store(v, dst);
}

extern "C" void kernel_launch(void* const* d_in, const int* in_sizes, int n_in,
                              void* d_out, int out_size, void* d_ws, size_t ws_size,
                              hipStream_t stream) {
    const float* proc_emb    = (const float*)d_in[0];
    const float* med_emb     = (const float*)d_in[1];
    const float* chart_emb   = (const float*)d_in[2];
    const int*   concept_ids = (const int*)d_in[3];
    const int*   token_type  = (const int*)d_in[4];
    float*       out         = (float*)d_out;

    const int threads = 256;                 // 8 wave32 per block
    const int blocks  = (N_TOKENS * 32) / threads;  // 4096, exact cover
    concept_embed_gather<<<blocks, threads, 0, stream>>>(
        proc_emb, med_emb, chart_emb, concept_ids, token_type, out);
}